// EG_GATLayer_15702400434747
// MI455X (gfx1250) — compile-verified
//
#include <hip/hip_runtime.h>
#include <math.h>

#define N_NODES 50000
#define N_EDGES 800000
#define IN_DIM 128
#define EDGE_DIM 16
#define N_HEADS 8
#define OUT_DIM 16
#define HD (N_HEADS * OUT_DIM) /* 128 */

typedef __attribute__((ext_vector_type(2))) float v2f;
typedef __attribute__((ext_vector_type(8))) float v8f;

// ---------------- float atomic max via int tricks ----------------
__device__ inline void atomicMaxFloat(float* addr, float v) {
  if (v >= 0.f) {
    atomicMax((int*)addr, __float_as_int(v));
  } else {
    atomicMin((unsigned int*)addr, (unsigned int)__float_as_int(v));
  }
}

// ---------------- init: zero out, m=-inf, denom=0 ----------------
__global__ void k_init(float* __restrict__ out, float* __restrict__ m,
                       float* __restrict__ denom) {
  int i = blockIdx.x * blockDim.x + threadIdx.x;
  if (i < N_NODES * HD) out[i] = 0.f;
  if (i < N_NODES * N_HEADS) {
    m[i] = -__builtin_inff();
    denom[i] = 0.f;
  }
}

// ---------------- node projection: h = x @ W_node (WMMA f32) ----------------
// Block: 256 threads (8 waves). Block covers 64 rows x 128 cols.
// Wave w -> cols [16w, 16w+16); loops over 4 row tiles of 16.
__global__ __launch_bounds__(256) void k_node_proj(
    const float* __restrict__ x, const float* __restrict__ Wn,
    float* __restrict__ h) {
  __shared__ float sW[IN_DIM * HD];  // [k][col], col = head*16 + d  (64 KB)
  for (int idx = threadIdx.x; idx < IN_DIM * HD; idx += 256) {
    int i = idx >> 7, col = idx & 127;
    sW[idx] = Wn[(col >> 4) * (IN_DIM * OUT_DIM) + i * OUT_DIM + (col & 15)];
  }
  __syncthreads();

  const int lane = threadIdx.x & 31;
  const int wave = threadIdx.x >> 5;
  const int l16 = lane & 15;
  const int hi = lane >> 4;
  const int col0 = wave * 16;
  const int rowBase = blockIdx.x * 64;

  for (int rt = 0; rt < 4; ++rt) {
    const int R = rowBase + rt * 16;
    int arow = R + l16;
    if (arow >= N_NODES) arow = N_NODES - 1;  // clamp; stores are guarded
    const float* xr = x + (size_t)arow * IN_DIM;

    v8f acc = {};
#pragma unroll
    for (int kk = 0; kk < IN_DIM / 4; ++kk) {
      const int k = kk * 4 + 2 * hi;
      v2f a;
      a.x = xr[k];
      a.y = xr[k + 1];
      v2f b;
      b.x = sW[(k + 0) * HD + col0 + l16];
      b.y = sW[(k + 1) * HD + col0 + l16];
      acc = __builtin_amdgcn_wmma_f32_16x16x4_f32(
          /*neg_a=*/false, a, /*neg_b=*/false, b,
          /*c_mod=*/(short)0, acc, /*reuse_a=*/false, /*reuse_b=*/false);
    }
#pragma unroll
    for (int r = 0; r < 8; ++r) {
      const int orow = R + r + 8 * hi;  // D layout: VGPR r -> M = r + 8*hi
      if (orow < N_NODES) h[(size_t)orow * HD + col0 + l16] = acc[r];
    }
  }
}

// ---------------- edge logits + segment max ----------------
// One edge per wave32. Lane l: head = l>>2, covers 4 of the 16 dims.
__global__ __launch_bounds__(256) void k_edge_logits(
    const float* __restrict__ h, const float* __restrict__ ea,
    const float* __restrict__ We, const int* __restrict__ src,
    const int* __restrict__ dst, float* __restrict__ logits,
    float* __restrict__ m) {
  __shared__ float sWsum[N_HEADS * EDGE_DIM];  // Wsum[h,i] = sum_d W_edge[h,i,d]
  if (threadIdx.x < N_HEADS * EDGE_DIM) {
    const int hh = threadIdx.x >> 4, i = threadIdx.x & 15;
    float s = 0.f;
    for (int d = 0; d < OUT_DIM; ++d)
      s += We[hh * (EDGE_DIM * OUT_DIM) + i * OUT_DIM + d];
    sWsum[threadIdx.x] = s;
  }
  __syncthreads();

  const int lane = threadIdx.x & 31;
  const int wave = threadIdx.x >> 5;
  const int e = blockIdx.x * 8 + wave;
  if (e >= N_EDGES) return;

  const int sn = src[e], dn = dst[e];
  const float4 q4 = ((const float4*)(h + (size_t)dn * HD))[lane];
  const float4 k4 = ((const float4*)(h + (size_t)sn * HD))[lane];
  float part = q4.x * k4.x + q4.y * k4.y + q4.z * k4.z + q4.w * k4.w;

  const int head = lane >> 2;
  const int seg = (lane & 3) * 4;
  const float4 ea4 = *(const float4*)(ea + (size_t)e * EDGE_DIM + seg);
  const float4 ws4 = *(const float4*)(sWsum + head * EDGE_DIM + seg);
  part += ea4.x * ws4.x + ea4.y * ws4.y + ea4.z * ws4.z + ea4.w * ws4.w;

  part += __shfl_xor(part, 1, 32);
  part += __shfl_xor(part, 2, 32);

  if ((lane & 3) == 0) {
    float v = part * 0.25f;            // scale = 1/sqrt(16)
    v = v > 0.f ? v : 0.2f * v;        // leaky_relu(0.2)
    logits[(size_t)e * N_HEADS + head] = v;
    atomicMaxFloat(&m[(size_t)dn * N_HEADS + head], v);
  }
}

// ---------------- exp + segment sum ----------------
__global__ void k_exp_denom(const float* __restrict__ logits,
                            const int* __restrict__ dst,
                            const float* __restrict__ m,
                            float* __restrict__ denom) {
  const int idx = blockIdx.x * blockDim.x + threadIdx.x;
  if (idx >= N_EDGES * N_HEADS) return;
  const int e = idx >> 3, hh = idx & 7;
  const int dn = dst[e];
  const float ex = expf(logits[idx] - m[(size_t)dn * N_HEADS + hh]);
  atomicAdd(&denom[(size_t)dn * N_HEADS + hh], ex);
}

// ---------------- gated message + scatter add ----------------
// One edge per wave32. Lane l -> cols 4l..4l+3 (head = l>>2, d = (l&3)*4 ..).
__global__ __launch_bounds__(256) void k_aggregate(
    const float* __restrict__ h, const float* __restrict__ ea,
    const float* __restrict__ We, const int* __restrict__ src,
    const int* __restrict__ dst, const float* __restrict__ logits,
    const float* __restrict__ m, const float* __restrict__ denom,
    float* __restrict__ out) {
  __shared__ float sWe[N_HEADS * EDGE_DIM * OUT_DIM];  // 8 KB
  for (int idx = threadIdx.x; idx < N_HEADS * EDGE_DIM * OUT_DIM; idx += 256)
    sWe[idx] = We[idx];
  __syncthreads();

  const int lane = threadIdx.x & 31;
  const int wave = threadIdx.x >> 5;
  const int e = blockIdx.x * 8 + wave;
  if (e >= N_EDGES) return;

  const int sn = src[e], dn = dst[e];
  const int head = lane >> 2;
  const int dbase = (lane & 3) * 4;

  // broadcast the 16 edge features through lanes 0..15 + shuffles
  const float eav = ea[(size_t)e * EDGE_DIM + (lane & 15)];
  float e0 = 0.f, e1 = 0.f, e2 = 0.f, e3 = 0.f;
#pragma unroll
  for (int i = 0; i < EDGE_DIM; ++i) {
    const float ai = __shfl(eav, i, 32);
    const float* w = sWe + head * (EDGE_DIM * OUT_DIM) + i * OUT_DIM + dbase;
    e0 += ai * w[0];
    e1 += ai * w[1];
    e2 += ai * w[2];
    e3 += ai * w[3];
  }

  const float lg = logits[(size_t)e * N_HEADS + head];
  const float mm = m[(size_t)dn * N_HEADS + head];
  const float dd = denom[(size_t)dn * N_HEADS + head];
  const float alpha = expf(lg - mm) / (dd + 1e-9f);

  const float4 k4 = *(const float4*)(h + (size_t)sn * HD + lane * 4);
  const float s0 = 1.f / (1.f + expf(-e0));
  const float s1 = 1.f / (1.f + expf(-e1));
  const float s2 = 1.f / (1.f + expf(-e2));
  const float s3 = 1.f / (1.f + expf(-e3));

  float* op = out + (size_t)dn * HD + lane * 4;
  atomicAdd(op + 0, k4.x * s0 * alpha);
  atomicAdd(op + 1, k4.y * s1 * alpha);
  atomicAdd(op + 2, k4.z * s2 * alpha);
  atomicAdd(op + 3, k4.w * s3 * alpha);
}

extern "C" void kernel_launch(void* const* d_in, const int* in_sizes, int n_in,
                              void* d_out, int out_size, void* d_ws,
                              size_t ws_size, hipStream_t stream) {
  const float* x = (const float*)d_in[0];
  const float* ea = (const float*)d_in[1];
  const float* Wn = (const float*)d_in[2];
  const float* We = (const float*)d_in[3];
  const int* src = (const int*)d_in[4];
  const int* dst = (const int*)d_in[5];
  float* out = (float*)d_out;

  // workspace layout (floats): h[6.4M] | logits[6.4M] | m[400k] | denom[400k]
  float* ws = (float*)d_ws;
  float* h = ws;
  float* logits = ws + (size_t)N_NODES * HD;                       // 6,400,000
  float* m = logits + (size_t)N_EDGES * N_HEADS;                   // +6,400,000
  float* denom = m + (size_t)N_NODES * N_HEADS;                    // +400,000

  k_init<<<(N_NODES * HD + 255) / 256, 256, 0, stream>>>(out, m, denom);
  k_node_proj<<<(N_NODES + 63) / 64, 256, 0, stream>>>(x, Wn, h);
  k_edge_logits<<<(N_EDGES + 7) / 8, 256, 0, stream>>>(h, ea, We, src, dst,
                                                       logits, m);
  k_exp_denom<<<(N_EDGES * N_HEADS + 255) / 256, 256, 0, stream>>>(logits, dst,
                                                                   m, denom);
  k_aggregate<<<(N_EDGES + 7) / 8, 256, 0, stream>>>(h, ea, We, src, dst,
                                                     logits, m, denom, out);
}